// TAGConv_26216480375292
// MI455X (gfx1250) — compile-verified
//
#include <hip/hip_runtime.h>
#include <hip/hip_bf16.h>

// ---------------------------------------------------------------------------
// TAGConv (K=2) for gfx1250 / MI455X.
//  N=100000 nodes, E=1.6M edges, D_IN=D_OUT=128, concat width 384.
//  Phase 1: degree via global_atomic_add_f32 (L2-side atomics; h planes fit
//           in the 192MB L2, so scatter traffic never round-trips HBM).
//  Phase 2: two edge-parallel SpMM hops, one wave per edge, float4 gather,
//           pre-scale by norm[src] fused into the gather, post-scale kernel.
//  Phase 3: fused [N,384]x[384,128] GEMM with V_WMMA_F32_16X16X4_F32
//           (full fp32 — matches the fp32 reference; GEMM is L2-bandwidth
//           bound so f32 WMMA rate is not the limiter).
// ---------------------------------------------------------------------------

#define D_FEAT   128
#define D_CAT    384
#define D_OUT    128
#define AS_STRIDE 388   // 384 + 4 pad: column reads hit 16 distinct LDS banks

typedef float v2f __attribute__((ext_vector_type(2)));
typedef float v8f __attribute__((ext_vector_type(8)));

// ---------------- utility: zero a float region ------------------------------
__global__ void k_zero(float* __restrict__ p, size_t n) {
    size_t i = (size_t)blockIdx.x * blockDim.x + threadIdx.x;
    if (i < n) p[i] = 0.0f;
}

// ---------------- degree: deg[dst[e]] += 1 ----------------------------------
__global__ void k_degree(const int* __restrict__ dst, float* __restrict__ deg,
                         int n_edges) {
    int e = blockIdx.x * blockDim.x + threadIdx.x;
    if (e < n_edges) atomicAdd(&deg[dst[e]], 1.0f);
}

// ---------------- norm[i] = deg[i]^-1/2 (in place) --------------------------
__global__ void k_rsqrt(float* __restrict__ d, int n) {
    int i = blockIdx.x * blockDim.x + threadIdx.x;
    if (i < n) d[i] = rsqrtf(d[i]);
}

// ---------------- SpMM hop: hout[dst] += norm[src] * hin[src] ---------------
// one wave (32 lanes) per edge; each lane owns a float4 column slice.
__global__ __launch_bounds__(256)
void k_spmm_scatter(const float* __restrict__ hin, const int* __restrict__ src,
                    const int* __restrict__ dst, const float* __restrict__ norm,
                    float* __restrict__ hout, int n_edges) {
    int wave = blockIdx.x * (blockDim.x >> 5) + (threadIdx.x >> 5);
    int lane = threadIdx.x & 31;
    if (wave >= n_edges) return;
    int s = src[wave];
    int d = dst[wave];
    float sc = norm[s];
    const float4 v = *(const float4*)(hin + (size_t)s * D_FEAT + lane * 4);
    float* op = hout + (size_t)d * D_FEAT + lane * 4;
    atomicAdd(op + 0, v.x * sc);
    atomicAdd(op + 1, v.y * sc);
    atomicAdd(op + 2, v.z * sc);
    atomicAdd(op + 3, v.w * sc);
}

// ---------------- post-scale: h[i][*] *= norm[i] ----------------------------
__global__ void k_postscale(float* __restrict__ h, const float* __restrict__ norm,
                            size_t total) {
    size_t i = (size_t)blockIdx.x * blockDim.x + threadIdx.x;
    if (i < total) h[i] *= norm[i >> 7];   // 128 columns per row
}

// ---------------- fused concat-GEMM with fp32 WMMA --------------------------
// out[m, n] = sum_k x[m,k] * W[n,k] + b[n], x = [h0 | h1 | h2]  (K = 384)
// block: 16 rows x 128 cols, 8 waves; wave w -> cols [16w, 16w+16).
// per wave: 96 chained v_wmma_f32_16x16x4_f32.
__global__ __launch_bounds__(256)
void k_gemm_wmma(const float* __restrict__ h0, const float* __restrict__ h1,
                 const float* __restrict__ h2, const float* __restrict__ W,
                 const float* __restrict__ bias, float* __restrict__ out,
                 int n_nodes) {
    __shared__ float As[16 * AS_STRIDE];            // 16 x 384 A tile (padded)

    const int m0  = blockIdx.x * 16;
    const int tid = threadIdx.x;

    // ---- stage the shared 16x384 A tile (concat) with coalesced float4 ----
    for (int i = tid; i < 16 * (D_CAT / 4); i += 256) {
        int row = i / (D_CAT / 4);
        int cc  = (i % (D_CAT / 4)) * 4;
        int node = m0 + row;
        if (node >= n_nodes) node = n_nodes - 1;    // clamp: keep EXEC full
        const float* sp;
        if (cc < D_FEAT)          sp = h0 + (size_t)node * D_FEAT + cc;
        else if (cc < 2 * D_FEAT) sp = h1 + (size_t)node * D_FEAT + (cc - D_FEAT);
        else                      sp = h2 + (size_t)node * D_FEAT + (cc - 2 * D_FEAT);
        *(float4*)&As[row * AS_STRIDE + cc] = *(const float4*)sp;
    }
    __syncthreads();

    const int wave  = tid >> 5;          // 0..7 -> N tile
    const int lane  = tid & 31;
    const int n0    = wave * 16;
    const int mrow  = lane & 15;         // M (for A) / N (for B) within tile
    const int khalf = (lane >> 4) * 2;   // lanes 16..31 hold K+2, K+3

    // B = W^T : lane reads W[n0+mrow][k] pairs (24KB/wave slice, L2-resident)
    const float* wrow = W + (size_t)(n0 + mrow) * D_CAT;
    const float* arow = &As[mrow * AS_STRIDE];

    v8f acc = {};
#pragma unroll 8
    for (int kb = 0; kb < D_CAT; kb += 4) {
        v2f a, b;
        a.x = arow[kb + khalf];
        a.y = arow[kb + khalf + 1];
        b.x = wrow[kb + khalf];
        b.y = wrow[kb + khalf + 1];
        // (neg_a, A, neg_b, B, c_mod, C, reuse_a, reuse_b)
        acc = __builtin_amdgcn_wmma_f32_16x16x4_f32(
                  false, a, false, b, (short)0, acc, false, false);
    }

    // ---- D layout: VGPR r -> (M = r + 8*(lane>=16), N = lane&15) ----
    const int mo   = (lane >> 4) * 8;
    const int ncol = n0 + (lane & 15);
    const float bv = bias[ncol];
#pragma unroll
    for (int r = 0; r < 8; ++r) {
        int m = m0 + r + mo;
        if (m < n_nodes)                              // predicate stores only
            out[(size_t)m * D_OUT + ncol] = acc[r] + bv;
    }
}

// ---------------------------------------------------------------------------
extern "C" void kernel_launch(void* const* d_in, const int* in_sizes, int n_in,
                              void* d_out, int out_size, void* d_ws, size_t ws_size,
                              hipStream_t stream) {
    const float* feat = (const float*)d_in[0];
    const int*   src  = (const int*)d_in[1];   // harness: integer -> int32
    const int*   dst  = (const int*)d_in[2];
    const float* W    = (const float*)d_in[3];
    const float* bias = (const float*)d_in[4];
    float*       out  = (float*)d_out;

    const int N = in_sizes[0] / D_FEAT;
    const int E = in_sizes[1];

    // workspace layout: [ norm (N, padded) | h1 (N*128) | h2 (N*128) ]
    const size_t normPad = ((size_t)N + 255) & ~(size_t)255;
    float* norm = (float*)d_ws;
    float* h1   = norm + normPad;
    float* h2   = h1 + (size_t)N * D_FEAT;
    const size_t zeroTotal = normPad + 2 * (size_t)N * D_FEAT;

    const size_t planeTotal = (size_t)N * D_FEAT;

    // 1) zero norm + h1 + h2
    {
        int blocks = (int)((zeroTotal + 255) / 256);
        k_zero<<<blocks, 256, 0, stream>>>((float*)d_ws, zeroTotal);
    }
    // 2) in-degree via fp32 atomics
    k_degree<<<(E + 255) / 256, 256, 0, stream>>>(dst, norm, E);
    // 3) norm = deg^-1/2
    k_rsqrt<<<(N + 255) / 256, 256, 0, stream>>>(norm, N);
    // 4) hop 1: h1 = scatter(norm[src]*feat[src]);  h1 *= norm
    k_spmm_scatter<<<(E + 7) / 8, 256, 0, stream>>>(feat, src, dst, norm, h1, E);
    k_postscale<<<(int)((planeTotal + 255) / 256), 256, 0, stream>>>(h1, norm, planeTotal);
    // 5) hop 2: h2 = scatter(norm[src]*h1[src]);   h2 *= norm
    k_spmm_scatter<<<(E + 7) / 8, 256, 0, stream>>>(h1, src, dst, norm, h2, E);
    k_postscale<<<(int)((planeTotal + 255) / 256), 256, 0, stream>>>(h2, norm, planeTotal);
    // 6) fused concat-GEMM + bias with fp32 WMMA
    k_gemm_wmma<<<(N + 15) / 16, 256, 0, stream>>>(feat, h1, h2, W, bias, out, N);
}